// FasterRCNN_24472723653114
// MI455X (gfx1250) — compile-verified
//
#include <hip/hip_runtime.h>
#include <cstdint>
#include <cstddef>

// FasterRCNN anchor->GT matching for MI455X (gfx1250, wave32).
//
// Per anchor: IoU against 64 GT boxes held in LDS, running (max, argmax) scan
// with bit-exact replication of the reference arithmetic (IEEE f32 div, no FMA
// contraction), then thresholded output of the matched GT row.
//
// CDNA5-specific paths used:
//   * global_load_async_to_lds_b128 + s_wait_asynccnt (async tensor/LDS DMA
//     path, ASYNCcnt) to stage the 1280 B GT table into LDS.
//   * wave32 codegen throughout; LDS broadcast reads (all lanes same address).

#define TPB 256   // 8 wave32s per block
#define APT 2     // anchors per thread
#define GTM 64    // GT boxes per batch (M)

#define IOU_LO 0.3f
#define IOU_HI 0.7f
#define NEUTRAL_VAL (-100000000.0f)

__global__ __launch_bounds__(TPB)
void faster_rcnn_match_kernel(const float* __restrict__ anchors,
                              const float* __restrict__ gt,
                              float* __restrict__ out,
                              int N)
{
#pragma clang fp contract(off)
    __shared__ float  raw[GTM * 5];   // async-DMA staging of gt rows
    __shared__ float4 gbox[GTM];      // x1,y1,x2,y2 (ds_load_b128 per iter)
    __shared__ float  gcls[GTM];      // class
    __shared__ float  garea[GTM];     // (x2-x1)*(y2-y1), reference op order

    const int tid = threadIdx.x;
    const int b   = blockIdx.y;
    const float* gts = gt + (size_t)b * (GTM * 5);

    // ---- CDNA5 async global -> LDS copy of the GT table (80 x b128 = 1280B)
    if (tid < (GTM * 5) / 4) {
        uint32_t lds_off = (uint32_t)(uintptr_t)(&raw[tid * 4]); // low 32b of
        uint64_t gaddr   = (uint64_t)(uintptr_t)(gts + tid * 4); // flat = LDS off
        asm volatile("global_load_async_to_lds_b128 %0, %1, off"
                     :: "v"(lds_off), "v"(gaddr) : "memory");
    }
    asm volatile("s_wait_asynccnt 0x0" ::: "memory");
    __syncthreads();

    // ---- preprocess GT rows into broadcast-friendly LDS layout
    if (tid < GTM) {
        float x1 = raw[tid * 5 + 0], y1 = raw[tid * 5 + 1];
        float x2 = raw[tid * 5 + 2], y2 = raw[tid * 5 + 3];
        gbox[tid]  = make_float4(x1, y1, x2, y2);
        gcls[tid]  = raw[tid * 5 + 4];
        garea[tid] = (x2 - x1) * (y2 - y1);
    }
    __syncthreads();

    const float4* a4   = (const float4*)anchors + (size_t)b * N;
    float*        outb = out + (size_t)b * N * 5;

    int    idx[APT];
    float4 av[APT];
    float  area1[APT];
    float  bq[APT];
    int    bidx[APT];

#pragma unroll
    for (int j = 0; j < APT; ++j) {
        idx[j]   = blockIdx.x * (TPB * APT) + j * TPB + tid;
        int ci   = idx[j] < N ? idx[j] : N - 1;     // clamp (discarded later)
        av[j]    = a4[ci];                          // global_load_b128
        area1[j] = (av[j].z - av[j].x) * (av[j].w - av[j].y);
        bq[j]    = -1.0f;                           // all IoU >= 0 -> first wins
        bidx[j]  = 0;
    }

    // ---- IoU max/argmax scan over 64 GT boxes (VALU-bound hot loop)
#pragma unroll 4
    for (int g = 0; g < GTM; ++g) {
        float4 gb = gbox[g];      // ds_load_b128, broadcast (conflict-free)
        float  a2 = garea[g];     // ds_load_b32,  broadcast
#pragma unroll
        for (int j = 0; j < APT; ++j) {
            float ltx = fmaxf(av[j].x, gb.x);
            float lty = fmaxf(av[j].y, gb.y);
            float rbx = fminf(av[j].z, gb.z);
            float rby = fminf(av[j].w, gb.w);
            float w   = fmaxf(rbx - ltx, 0.0f);
            float h   = fmaxf(rby - lty, 0.0f);
            float inter = w * h;
            float s   = area1[j] + a2;   // reference order: area1 + area2
            float u   = s - inter;       // contract(off): no fma(-w,h,s)
            float d   = u + 1e-6f;
            float q   = inter / d;       // IEEE-correct f32 div (div_fmas seq)
            bool better = q > bq[j];     // strict '>' == argmax-first semantics
            bq[j]   = better ? q     : bq[j];
            bidx[j] = better ? g     : bidx[j];
        }
    }

    // ---- gather matched GT row, apply thresholds, store 5 floats per anchor
#pragma unroll
    for (int j = 0; j < APT; ++j) {
        if (idx[j] >= N) continue;
        float4 mb = gbox[bidx[j]];     // divergent LDS gather
        float  mc = gcls[bidx[j]];
        float  q  = bq[j];
        float o0 = mb.x, o1 = mb.y, o2 = mb.z, o3 = mb.w, o4 = mc;
        if (q <= IOU_LO) {
            o0 = o1 = o2 = o3 = o4 = -1.0f;
        } else if (q < IOU_HI) {
            o0 = o1 = o2 = o3 = o4 = NEUTRAL_VAL;
        }
        float* p = outb + (size_t)idx[j] * 5;
        p[0] = o0; p[1] = o1; p[2] = o2; p[3] = o3; p[4] = o4;
    }
}

extern "C" void kernel_launch(void* const* d_in, const int* in_sizes, int n_in,
                              void* d_out, int out_size, void* d_ws, size_t ws_size,
                              hipStream_t stream) {
    (void)n_in; (void)out_size; (void)d_ws; (void)ws_size;
    const float* anchors = (const float*)d_in[0];   // (B, N, 4) f32
    const float* gt      = (const float*)d_in[1];   // (B, 64, 5) f32
    float*       out     = (float*)d_out;           // (B, N, 5) f32

    const int M = GTM;
    int B = in_sizes[1] / (M * 5);        // 8
    if (B < 1) B = 1;
    int N = in_sizes[0] / (B * 4);        // 200000

    dim3 grid((N + TPB * APT - 1) / (TPB * APT), B, 1);
    faster_rcnn_match_kernel<<<grid, dim3(TPB, 1, 1), 0, stream>>>(anchors, gt, out, N);
}